// PYG_GCN_15118284881960
// MI455X (gfx1250) — compile-verified
//
#include <hip/hip_runtime.h>

#define NN 50000
#define NE 600000
#define D  128
#define NCLS 40

typedef __attribute__((ext_vector_type(2))) float v2f;
typedef __attribute__((ext_vector_type(8))) float v8f;

// ---------------------------------------------------------------- degrees
__global__ void deg_init_kernel(float* __restrict__ deg) {
    int i = blockIdx.x * blockDim.x + threadIdx.x;
    if (i < NN) deg[i] = 1.0f;   // self-loop contributes 1 to in-degree
}

__global__ void deg_count_kernel(const long long* __restrict__ ei,
                                 float* __restrict__ deg) {
    int e = blockIdx.x * blockDim.x + threadIdx.x;
    if (e < NE) {
        int c = (int)ei[NE + e];          // col = edge_index[1]
        unsafeAtomicAdd(&deg[c], 1.0f);
    }
}

__global__ void dinv_kernel(const float* __restrict__ deg,
                            float* __restrict__ dinv) {
    int i = blockIdx.x * blockDim.x + threadIdx.x;
    if (i < NN) {
        float d = deg[i];
        dinv[i] = (d > 0.0f) ? rsqrtf(d) : 0.0f;
    }
}

// ---------------------------------------------------------------- WMMA GEMM
// out[NN x NCOLS] = op(A[NN x 128]) @ W[128 x NCOLS] (+obias) (optional relu)
// PRE:   A element -> relu(A + abias[k]) applied while staging to LDS
// Block: 128 threads = 4 waves; each wave owns a 16-row tile, NT col-tiles.
// W is staged TRANSPOSED (K-contiguous) so B fragments are single ds_load_b64.
template <int NT, int NCOLS, bool PRE, bool RELUO, bool BIASO>
__launch_bounds__(128)
__global__ void gemm_wmma_kernel(const float* __restrict__ A,
                                 const float* __restrict__ W,
                                 const float* __restrict__ abias,
                                 const float* __restrict__ obias,
                                 float* __restrict__ out) {
    constexpr int AS_STRIDE = D + 4;   // A row pad: conflict-free frag reads
    constexpr int WT_STRIDE = 36;      // K-stride pad (32+4): frag reads clean
    __shared__ float As[4][16 * AS_STRIDE];       // 33,792 B
    __shared__ float WsT[NT * 16 * WT_STRIDE];    // NT=8: 18,432 B

    const int tid  = threadIdx.x;
    const int wave = tid >> 5;
    const int lane = tid & 31;
    const int rowTile = blockIdx.x * 4 + wave;
    const bool valid  = rowTile < (NN / 16);
    const int row0    = rowTile * 16;

    // ---- stage 16x128 A tile per wave (coalesced float4 loads)
    if (valid) {
        for (int idx = lane; idx < 16 * 32; idx += 32) {
            const int r  = idx >> 5;
            const int c4 = (idx & 31) << 2;
            float4 v = *(const float4*)(A + (size_t)(row0 + r) * D + c4);
            if (PRE) {
                v.x = fmaxf(v.x + abias[c4 + 0], 0.0f);
                v.y = fmaxf(v.y + abias[c4 + 1], 0.0f);
                v.z = fmaxf(v.z + abias[c4 + 2], 0.0f);
                v.w = fmaxf(v.w + abias[c4 + 3], 0.0f);
            }
            float* dst = &As[wave][r * AS_STRIDE + c4];
            dst[0] = v.x; dst[1] = v.y; dst[2] = v.z; dst[3] = v.w;
        }
    }

    v8f acc[NT];
#pragma unroll
    for (int n = 0; n < NT; ++n) { v8f z = {}; acc[n] = z; }

    const int m     = lane & 15;          // row (A) / col (B) within tile
    const int khalf = (lane >> 4) << 1;   // 0 for lanes 0-15, 2 for 16-31

    for (int kc = 0; kc < D; kc += 32) {
        // ---- stage W[kc..kc+32) x NCOLS transposed: WsT[c][r] (zero-pad cols)
        for (int idx = tid; idx < 32 * NT * 16; idx += 128) {
            const int r = idx / (NT * 16);      // k within chunk (coalesced read)
            const int c = idx % (NT * 16);      // output column
            WsT[c * WT_STRIDE + r] =
                (c < NCOLS) ? W[(size_t)(kc + r) * NCOLS + c] : 0.0f;
        }
        __syncthreads();

#pragma unroll
        for (int k8 = 0; k8 < 8; ++k8) {
            const int k = kc + k8 * 4;
            // A frag: 2 contiguous floats -> one ds_load_b64
            const v2f a = *(const v2f*)&As[wave][m * AS_STRIDE + k + khalf];
#pragma unroll
            for (int n = 0; n < NT; ++n) {
                // B frag: K-contiguous in WsT -> one ds_load_b64, no repacking
                const v2f b =
                    *(const v2f*)&WsT[(n * 16 + m) * WT_STRIDE + k8 * 4 + khalf];
                acc[n] = __builtin_amdgcn_wmma_f32_16x16x4_f32(
                    false, a, false, b, (short)0, acc[n], false, false);
            }
        }
        __syncthreads();
    }

    if (valid) {
        const int rbase = row0 + ((lane >> 4) << 3);
#pragma unroll
        for (int n = 0; n < NT; ++n) {
            const int col = n * 16 + m;
            if (col < NCOLS) {
                const float bo = BIASO ? obias[col] : 0.0f;
#pragma unroll
                for (int v = 0; v < 8; ++v) {
                    float y = acc[n][v] + bo;
                    if (RELUO) y = fmaxf(y, 0.0f);
                    out[(size_t)(rbase + v) * NCOLS + col] = y;
                }
            }
        }
    }
}

// ---------------------------------------------------------------- self-loop init
// g[i,:] = dinv[i]^2 * xw[i,:]
__global__ void selfloop_kernel(const float* __restrict__ xw,
                                const float* __restrict__ dinv,
                                float* __restrict__ g) {
    size_t i = (size_t)blockIdx.x * blockDim.x + threadIdx.x; // over N*D/4
    if (i < (size_t)NN * (D / 4)) {
        const size_t node = i / (D / 4);
        float s = dinv[node];
        s = s * s;
        float4 v = ((const float4*)xw)[i];
        float4 o;
        o.x = v.x * s; o.y = v.y * s; o.z = v.z * s; o.w = v.w * s;
        ((float4*)g)[i] = o;
    }
}

// ---------------------------------------------------------------- edge scatter
// one wave32 per edge: g[col,:] += dinv[row]*dinv[col] * xw[row,:]
__global__ void scatter_kernel(const long long* __restrict__ ei,
                               const float* __restrict__ dinv,
                               const float* __restrict__ xw,
                               float* __restrict__ g) {
    const int lane = threadIdx.x & 31;
    const int wid  = (blockIdx.x * blockDim.x + threadIdx.x) >> 5;
    const int nw   = (gridDim.x * blockDim.x) >> 5;
    for (int e = wid; e < NE; e += nw) {
        const int r = (int)ei[e];
        const int c = (int)ei[NE + e];
        const float nrm = dinv[r] * dinv[c];
        const float4 v = ((const float4*)(xw + (size_t)r * D))[lane];
        float* dst = g + (size_t)c * D + lane * 4;
        unsafeAtomicAdd(dst + 0, v.x * nrm);   // global_atomic_add_f32, L2-resident
        unsafeAtomicAdd(dst + 1, v.y * nrm);
        unsafeAtomicAdd(dst + 2, v.z * nrm);
        unsafeAtomicAdd(dst + 3, v.w * nrm);
    }
}

// ---------------------------------------------------------------- launch
extern "C" void kernel_launch(void* const* d_in, const int* in_sizes, int n_in,
                              void* d_out, int out_size, void* d_ws, size_t ws_size,
                              hipStream_t stream) {
    const float*     x     = (const float*)d_in[0];
    const long long* ei    = (const long long*)d_in[1];   // int64 [2, NE]
    const float*     W_in  = (const float*)d_in[2];
    const float*     b_in  = (const float*)d_in[3];
    const float*     W_gcn = (const float*)d_in[4];
    const float*     b_gcn = (const float*)d_in[5];
    const float*     W_cls = (const float*)d_in[6];
    const float*     b_cls = (const float*)d_in[7];
    float*           out   = (float*)d_out;

    float* ws   = (float*)d_ws;
    float* deg  = ws;                                  // [NN]
    float* dinv = ws + NN;                             // [NN]
    float* hbuf = ws + 2 * (size_t)NN;                 // [NN*D]  h, reused as g
    float* xw   = hbuf + (size_t)NN * D;               // [NN*D]

    deg_init_kernel <<<(NN + 255) / 256, 256, 0, stream>>>(deg);
    deg_count_kernel<<<(NE + 255) / 256, 256, 0, stream>>>(ei, deg);
    dinv_kernel     <<<(NN + 255) / 256, 256, 0, stream>>>(deg, dinv);

    const int gemmBlocks = (NN / 16 + 3) / 4;   // 782 blocks x 4 waves x 16 rows

    // h = relu(x @ W_in + b_in)
    gemm_wmma_kernel<8, D, false, true, true>
        <<<gemmBlocks, 128, 0, stream>>>(x, W_in, nullptr, b_in, hbuf);
    // xw = h @ W_gcn
    gemm_wmma_kernel<8, D, false, false, false>
        <<<gemmBlocks, 128, 0, stream>>>(hbuf, W_gcn, nullptr, nullptr, xw);
    // g (over hbuf): self-loop term, then edge scatter-add
    selfloop_kernel<<<((NN * (D / 4)) + 255) / 256, 256, 0, stream>>>(xw, dinv, hbuf);
    scatter_kernel <<<(NE * 32) / 256, 256, 0, stream>>>(ei, dinv, xw, hbuf);
    // y = relu(g + b_gcn) @ W_cls + b_cls   (pre-bias+relu fused into A staging)
    gemm_wmma_kernel<3, NCLS, true, false, true>
        <<<gemmBlocks, 128, 0, stream>>>(hbuf, W_cls, b_gcn, b_cls, out);
}